// MultiHeadAttention_10866267259030
// MI455X (gfx1250) — compile-verified
//
#include <hip/hip_runtime.h>
#include <hip/hip_bf16.h>

typedef __attribute__((ext_vector_type(2))) float v2f;
typedef __attribute__((ext_vector_type(8))) float v8f;

#define D_MODEL 1024
#define NH 16
#define DH 64
#define SEQ 16
#define BATCH 8
#define LC 8192
#define LTOT (LC + SEQ)                     // 8208
#define NTILES (LTOT / 16)                  // 513
#define KVSTRIDE (BATCH * NH * DH)          // 8192 floats per time step
#define RES_ELEMS (SEQ * BATCH * D_MODEL)   // 131072
#define KV_ELEMS ((size_t)LTOT * (size_t)KVSTRIDE)  // 67239936
#define NW 4                                 // waves per attention block
#define PARTIAL_STRIDE (2 * SEQ + SEQ * DH)  // 16 m + 16 l + 1024 O = 1056 floats

// D = A(16x4) * B(4x16) + C, fp32 WMMA.
// A: lanes 0-15 hold {K=0,K=1} of row M=lane; lanes 16-31 hold {K=2,K=3}.
// B: lanes 0-15 hold {K=0,K=1} of col N=lane; lanes 16-31 hold {K=2,K=3}.
// C/D: VGPR i: lanes 0-15 -> D[i][lane], lanes 16-31 -> D[i+8][lane-16].
__device__ __forceinline__ v8f wmma4(v2f a, v2f b, v8f c) {
  return __builtin_amdgcn_wmma_f32_16x16x4_f32(
      /*neg_a=*/false, a, /*neg_b=*/false, b,
      /*c_mod=*/(short)0, c, /*reuse_a=*/false, /*reuse_b=*/false);
}

// ---------------------------------------------------------------------------
// QKV projection: Y = X(128x1024) @ W(1024x1024).
// grid = (64, 8, 3): 16x16 output tile per wave, z selects Wq/Wk/Wv.
// Q -> workspace [b][h][s][d]; K,V -> appended rows of new_k/new_v outputs.
// ---------------------------------------------------------------------------
__global__ __launch_bounds__(32) void qkv_proj_kernel(
    const float* __restrict__ X,
    const float* __restrict__ Wq, const float* __restrict__ Wk,
    const float* __restrict__ Wv,
    float* __restrict__ qws, float* __restrict__ outk, float* __restrict__ outv) {
  const int lane = threadIdx.x;
  const int lo = lane & 15, hi = lane >> 4;
  const int n0 = blockIdx.x << 4;
  const int m0 = blockIdx.y << 4;
  const int mode = blockIdx.z;
  const float* W = (mode == 0) ? Wq : (mode == 1 ? Wk : Wv);

  const float* xrow = X + (size_t)(m0 + lo) * D_MODEL + 2 * hi;
  const float* wcol = W + (size_t)(2 * hi) * D_MODEL + n0 + lo;

  v8f acc = {};
#pragma unroll 8
  for (int kk = 0; kk < D_MODEL; kk += 4) {
    v2f a;
    a.x = xrow[kk];
    a.y = xrow[kk + 1];
    v2f b;
    b.x = wcol[(size_t)kk * D_MODEL];
    b.y = wcol[(size_t)(kk + 1) * D_MODEL];
    acc = wmma4(a, b, acc);
  }

#pragma unroll
  for (int i = 0; i < 8; ++i) {
    const int r = m0 + i + 8 * hi;  // row in (S*B)
    const int c = n0 + lo;          // col in (H*DH)
    const int s = r >> 3, b = r & 7;
    const float val = acc[i];
    if (mode == 0) {
      const int h = c >> 6, d = c & 63;
      qws[(((size_t)b * NH + h) * SEQ + s) * DH + d] = val;
    } else {
      float* dst = (mode == 1) ? outk : outv;
      dst[(size_t)(LC + s) * KVSTRIDE + (size_t)b * D_MODEL + c] = val;
    }
  }
}

// ---------------------------------------------------------------------------
// Fused flash-attention partial + cache write-through.
// grid = (NSPLIT, 128): blockIdx.y = (b,h); blockIdx.x = L-split.
// Each of the NW waves handles tiles j = blockIdx.x*NW + w, step gridDim.x*NW.
// K/V tiles in the cache region are read from cache_k/cache_v and *stored*
// to new_k/new_v while feeding the WMMAs (each element is touched exactly
// once), so the 537 MB cache copy costs no extra HBM reads. The appended tile
// (keys >= LC) is read from new_k/new_v (written by qkv_proj).
// Per-block partial (m, l, O) is LDS-merged across waves and written to ws.
// ---------------------------------------------------------------------------
__global__ __launch_bounds__(32 * NW) void attn_partial_kernel(
    const float* __restrict__ qws,
    const float* __restrict__ cache_k, const float* __restrict__ cache_v,
    float* __restrict__ outk, float* __restrict__ outv,
    float* __restrict__ pws) {
  __shared__ float lds_p[NW][16][16];
  __shared__ float lds_o[NW][16][DH];
  __shared__ float lds_m[NW][16];
  __shared__ float lds_l[NW][16];

  const int tid = threadIdx.x;
  const int lane = tid & 31;
  const int w = tid >> 5;
  const int lo = lane & 15, hi = lane >> 4;
  const int bh = blockIdx.y;  // b*NH + h
  const int b = bh >> 4, h = bh & 15;

  const size_t boff = (size_t)b * D_MODEL + h * DH;  // within one t-slice

  // Q (16x64) into A-operand registers, 1/sqrt(64) scale folded in.
  v2f qa[16];
  {
    const float* qrow = qws + (size_t)bh * (SEQ * DH) + lo * DH + 2 * hi;
#pragma unroll
    for (int t = 0; t < 16; ++t) {
      qa[t].x = qrow[4 * t] * 0.125f;
      qa[t].y = qrow[4 * t + 1] * 0.125f;
    }
  }

  float m[8], l[8];
  v8f oacc[4] = {v8f{}, v8f{}, v8f{}, v8f{}};
#pragma unroll
  for (int i = 0; i < 8; ++i) {
    m[i] = -1e30f;
    l[i] = 0.f;
  }

  const int stride = gridDim.x * NW;  // wave-slots per (b,h)
  for (int j = blockIdx.x * NW + w; j < NTILES; j += stride) {
    const int t0 = j << 4;
    const bool incache = (t0 < LC);

    // prefetch this wave's next tile (K and V rows; 256B = 2 lines per row)
    {
      const int jn = j + stride;
      if (jn < (LC / 16)) {
        const size_t poff = (size_t)((jn << 4) + (lane >> 1)) * KVSTRIDE +
                            boff + (size_t)(lane & 1) * 32;
        __builtin_prefetch(cache_k + poff, 0, 1);
        __builtin_prefetch(cache_v + poff, 0, 1);
      }
    }

    // ---- scores tile (16 queries x 16 keys) = Q @ K^T, with write-through
    const size_t kroff = (size_t)(t0 + lo) * KVSTRIDE + boff + 2 * hi;
    const float* kr = (incache ? cache_k : outk) + kroff;
    float* kw = outk + kroff;
    v8f sc = {};
#pragma unroll
    for (int t = 0; t < 16; ++t) {
      const v2f kb = *(const v2f*)(kr + 4 * t);
      if (incache) *(v2f*)(kw + 4 * t) = kb;
      sc = wmma4(qa[t], kb, sc);
    }

    // causal mask: only the final tile (keys LC..LC+15) is partially masked
    if (!incache) {
#pragma unroll
      for (int i = 0; i < 8; ++i) {
        if ((t0 + lo - LC) > (i + 8 * hi)) sc[i] = -1e30f;
      }
    }

    // ---- online softmax; row stats uniform within each 16-lane half
    float p[8];
#pragma unroll
    for (int i = 0; i < 8; ++i) {
      float tm = sc[i];
      tm = fmaxf(tm, __shfl_xor(tm, 1, 32));
      tm = fmaxf(tm, __shfl_xor(tm, 2, 32));
      tm = fmaxf(tm, __shfl_xor(tm, 4, 32));
      tm = fmaxf(tm, __shfl_xor(tm, 8, 32));
      const float nm = fmaxf(m[i], tm);
      const float corr = __expf(m[i] - nm);
      const float pv = __expf(sc[i] - nm);
      float ps = pv;
      ps += __shfl_xor(ps, 1, 32);
      ps += __shfl_xor(ps, 2, 32);
      ps += __shfl_xor(ps, 4, 32);
      ps += __shfl_xor(ps, 8, 32);
      m[i] = nm;
      l[i] = l[i] * corr + ps;
#pragma unroll
      for (int dc = 0; dc < 4; ++dc) oacc[dc][i] *= corr;
      p[i] = pv;
    }

    // re-layout P (C-matrix form -> A-matrix form) through LDS
#pragma unroll
    for (int i = 0; i < 8; ++i) lds_p[w][i + 8 * hi][lo] = p[i];

    // ---- O(16x64) += P(16x16) @ V(16x64), with write-through
#pragma unroll
    for (int kk = 0; kk < 16; kk += 4) {
      v2f a;
      a.x = lds_p[w][lo][kk + 2 * hi];
      a.y = lds_p[w][lo][kk + 2 * hi + 1];
      const size_t voff = (size_t)(t0 + kk + 2 * hi) * KVSTRIDE + boff;
      const float* vr = (incache ? cache_v : outv) + voff;
      float* vw = outv + voff;
#pragma unroll
      for (int dc = 0; dc < 4; ++dc) {
        v2f bb;
        bb.x = vr[dc * 16 + lo];
        bb.y = vr[KVSTRIDE + dc * 16 + lo];
        if (incache) {
          vw[dc * 16 + lo] = bb.x;
          vw[KVSTRIDE + dc * 16 + lo] = bb.y;
        }
        oacc[dc] = wmma4(a, bb, oacc[dc]);
      }
    }
  }

  // ---- publish per-wave partial (m, l, O) and merge across the NW waves
  if (lo == 0) {
#pragma unroll
    for (int i = 0; i < 8; ++i) {
      lds_m[w][i + 8 * hi] = m[i];
      lds_l[w][i + 8 * hi] = l[i];
    }
  }
#pragma unroll
  for (int i = 0; i < 8; ++i) {
#pragma unroll
    for (int dc = 0; dc < 4; ++dc)
      lds_o[w][i + 8 * hi][dc * 16 + lo] = oacc[dc][i];
  }
  __syncthreads();

  float* pb = pws + ((size_t)bh * gridDim.x + blockIdx.x) * PARTIAL_STRIDE;
  if (w == 0) {
    if (lane < 16) {
      const int r = lane;
      float M = -1e30f;
      for (int ww = 0; ww < NW; ++ww) M = fmaxf(M, lds_m[ww][r]);
      float Ls = 0.f;
      for (int ww = 0; ww < NW; ++ww)
        Ls += lds_l[ww][r] * __expf(lds_m[ww][r] - M);
      pb[r] = M;
      pb[16 + r] = Ls;
    }
    for (int idx = lane; idx < SEQ * DH; idx += 32) {
      const int r = idx >> 6;
      float M = -1e30f;
      for (int ww = 0; ww < NW; ++ww) M = fmaxf(M, lds_m[ww][r]);
      float Os = 0.f;
      for (int ww = 0; ww < NW; ++ww)
        Os += lds_o[ww][r][idx & 63] * __expf(lds_m[ww][r] - M);
      pb[32 + idx] = Os;
    }
  }
}

// ---------------------------------------------------------------------------
// Combine the nsplit L-partials per (b,h): final softmax normalization.
// ---------------------------------------------------------------------------
__global__ __launch_bounds__(64) void attn_combine_kernel(
    const float* __restrict__ pws, float* __restrict__ ctxws, int nsplit) {
  const int bh = blockIdx.x;
  const int b = bh >> 4, h = bh & 15;
  const float* pb0 = pws + (size_t)bh * nsplit * PARTIAL_STRIDE;
  for (int idx = threadIdx.x; idx < SEQ * DH; idx += 64) {
    const int r = idx >> 6, d = idx & 63;
    float M = -1e30f;
    for (int sp = 0; sp < nsplit; ++sp)
      M = fmaxf(M, pb0[(size_t)sp * PARTIAL_STRIDE + r]);
    float Ls = 0.f, Os = 0.f;
    for (int sp = 0; sp < nsplit; ++sp) {
      const float* pb = pb0 + (size_t)sp * PARTIAL_STRIDE;
      const float e = __expf(pb[r] - M);
      Ls += pb[16 + r] * e;
      Os += pb[32 + idx] * e;
    }
    ctxws[((size_t)r * BATCH + b) * D_MODEL + h * DH + d] = Os / Ls;
  }
}

// ---------------------------------------------------------------------------
// Output projection: res = CTX(128x1024) @ Wo(1024x1024)
// ---------------------------------------------------------------------------
__global__ __launch_bounds__(32) void out_proj_kernel(
    const float* __restrict__ A, const float* __restrict__ W,
    float* __restrict__ C) {
  const int lane = threadIdx.x;
  const int lo = lane & 15, hi = lane >> 4;
  const int n0 = blockIdx.x << 4;
  const int m0 = blockIdx.y << 4;

  const float* arow = A + (size_t)(m0 + lo) * D_MODEL + 2 * hi;
  const float* wcol = W + (size_t)(2 * hi) * D_MODEL + n0 + lo;

  v8f acc = {};
#pragma unroll 8
  for (int kk = 0; kk < D_MODEL; kk += 4) {
    v2f a;
    a.x = arow[kk];
    a.y = arow[kk + 1];
    v2f b;
    b.x = wcol[(size_t)kk * D_MODEL];
    b.y = wcol[(size_t)(kk + 1) * D_MODEL];
    acc = wmma4(a, b, acc);
  }

#pragma unroll
  for (int i = 0; i < 8; ++i) {
    const int r = m0 + i + 8 * hi;
    C[(size_t)r * D_MODEL + n0 + lo] = acc[i];
  }
}

extern "C" void kernel_launch(void* const* d_in, const int* in_sizes, int n_in,
                              void* d_out, int out_size, void* d_ws,
                              size_t ws_size, hipStream_t stream) {
  (void)in_sizes;
  (void)n_in;
  (void)out_size;

  const float* x = (const float*)d_in[0];
  const float* cache_k = (const float*)d_in[1];
  const float* cache_v = (const float*)d_in[2];
  const float* Wq = (const float*)d_in[3];
  const float* Wk = (const float*)d_in[4];
  const float* Wv = (const float*)d_in[5];
  const float* Wo = (const float*)d_in[6];

  float* res = (float*)d_out;
  float* outk = res + RES_ELEMS;
  float* outv = outk + KV_ELEMS;

  float* qws = (float*)d_ws;        // 128x1024 floats
  float* ctxws = qws + RES_ELEMS;   // 128x1024 floats
  float* pws = ctxws + RES_ELEMS;   // nsplit partials per (b,h)

  // deterministic split selection from available scratch
  int nsplit = 32;
  while (nsplit > 1 &&
         ((size_t)2 * RES_ELEMS +
          (size_t)BATCH * NH * nsplit * PARTIAL_STRIDE) * sizeof(float) >
             ws_size)
    nsplit >>= 1;

  dim3 pg(D_MODEL / 16, (SEQ * BATCH) / 16, 3);
  qkv_proj_kernel<<<pg, 32, 0, stream>>>(x, Wq, Wk, Wv, qws, outk, outv);

  dim3 ag(nsplit, BATCH * NH);
  attn_partial_kernel<<<ag, 32 * NW, 0, stream>>>(qws, cache_k, cache_v, outk,
                                                  outv, pws);

  attn_combine_kernel<<<BATCH * NH, 64, 0, stream>>>(pws, ctxws, nsplit);

  dim3 og(D_MODEL / 16, (SEQ * BATCH) / 16);
  out_proj_kernel<<<og, 32, 0, stream>>>(ctxws, Wo, res);
}